// FundamentalConsensus_36318243455431
// MI455X (gfx1250) — compile-verified
//
#include <hip/hip_runtime.h>
#include <hip/hip_bf16.h>
#include <math.h>

// ---------------------------------------------------------------------------
// PointNet binseg + mutual-NN matching for MI455X (gfx1250), fp32 WMMA path.
// Heavy GEMMs (match 4096x4096x256, all 1x1-conv layers, T2 apply) use
// V_WMMA_F32_16X16X4_F32 with 64x64 tiles, 4 waves/block, and double-buffered
// LDS K-panels filled by GLOBAL_LOAD_ASYNC_TO_LDS_B128 (ASYNCcnt-tracked DMA,
// overlapped with the WMMA burst).  Distance argmins are fused into the match
// GEMM epilogue via packed u64 global_atomic_min (order-independent).
// ---------------------------------------------------------------------------

typedef float v2f __attribute__((ext_vector_type(2)));
typedef float v8f __attribute__((ext_vector_type(8)));

#define NPTS  4096
#define DFEAT 256
#define BATCH 8
#define LDSTR 68          // padded LDS row stride (floats) for [32][64] panels

__device__ __forceinline__ v8f wmma4(v2f a, v2f b, v8f c) {
  // V_WMMA_F32_16X16X4_F32 : D = A(16x4) * B(4x16) + C(16x16), fp32
  return __builtin_amdgcn_wmma_f32_16x16x4_f32(false, a, false, b, (short)0, c,
                                               false, false);
}

__device__ __forceinline__ unsigned int ordf(float f) {
  unsigned int u = __float_as_uint(f);
  return (u & 0x80000000u) ? ~u : (u | 0x80000000u);  // monotonic float->uint
}

// LDS offset of a __shared__ object: flat LDS-aperture addresses carry the
// wave-relative LDS byte offset in addr[31:0] (ISA 10.2 aperture mapping).
__device__ __forceinline__ unsigned lds_off(const void* p) {
  return (unsigned)(unsigned long long)p;
}

// Async DMA: global -> LDS, 16 bytes per lane, tracked with ASYNCcnt.
#define ASYNC_B128(dst_lds, gaddr)                                            \
  asm volatile("global_load_async_to_lds_b128 %0, %1, off"                    \
               :: "v"(dst_lds), "v"(gaddr) : "memory")
#define WAIT_ASYNC0() asm volatile("s_wait_asynccnt 0" ::: "memory")

// ---------------------------------------------------------------------------
// Stage 1: per-point squared norms
// ---------------------------------------------------------------------------
__global__ void norms_kernel(const float* __restrict__ F, float* __restrict__ out) {
  int b = blockIdx.y;
  int n = blockIdx.x * 256 + threadIdx.x;
  const float* p = F + (size_t)b * DFEAT * NPTS + n;
  float s = 0.f;
  for (int d = 0; d < DFEAT; ++d) { float v = p[(size_t)d * NPTS]; s += v * v; }
  out[b * NPTS + n] = s;
}

__global__ void init_u64_kernel(unsigned long long* p, int n) {
  int i = blockIdx.x * 256 + threadIdx.x;
  if (i < n) p[i] = 0xFFFFFFFFFFFFFFFFULL;
}

// ---------------------------------------------------------------------------
// Stage 2: fused cross-distance GEMM + row/col argmin.
// Block = 128 threads (4 waves), computes a 64x64 tile of D[b] = a2+b2-2*A^T B.
// Wave w owns rows [16w,16w+16) x 64 cols = 4 WMMA subtiles.
// K-panels (32 x 64) double-buffered in LDS, filled by async DMA.
// ---------------------------------------------------------------------------
__global__ __launch_bounds__(128) void match_kernel(
    const float* __restrict__ AF, const float* __restrict__ BF,
    const float* __restrict__ a2, const float* __restrict__ b2,
    unsigned long long* __restrict__ rowmin,
    unsigned long long* __restrict__ colmin) {
  const int b  = blockIdx.z;
  const int j0 = blockIdx.x * 64;
  const int i0 = blockIdx.y * 64;
  const float* Ab = AF + (size_t)b * DFEAT * NPTS;
  const float* Bb = BF + (size_t)b * DFEAT * NPTS;

  __shared__ __align__(16) float As[2][32 * LDSTR];
  __shared__ __align__(16) float Bs[2][32 * LDSTR];
  __shared__ float Dt[64 * 65];

  const int tid  = threadIdx.x;
  const int lane = tid & 31;
  const int wave = tid >> 5;      // 0..3 -> row subtile
  const int l16  = lane & 15;
  const int hi   = lane >> 4;     // lane half selects K pair / M+8

  auto issue_panels = [&](int k0, int buf) {
    for (int r = 0; r < 4; ++r) {               // 512 float4 per matrix
      int idx = tid + r * 128;
      int row = idx >> 4;
      int c4  = (idx & 15) << 2;
      ASYNC_B128(lds_off(&As[buf][row * LDSTR + c4]),
                 (unsigned long long)(Ab + (size_t)(k0 + row) * NPTS + i0 + c4));
      ASYNC_B128(lds_off(&Bs[buf][row * LDSTR + c4]),
                 (unsigned long long)(Bb + (size_t)(k0 + row) * NPTS + j0 + c4));
    }
  };

  v8f acc[4];
  for (int s = 0; s < 4; ++s)
    for (int q = 0; q < 8; ++q) acc[s][q] = 0.f;

  issue_panels(0, 0);
  WAIT_ASYNC0();
  __syncthreads();

  for (int k0 = 0; k0 < DFEAT; k0 += 32) {
    int cur = (k0 >> 5) & 1;
    if (k0 + 32 < DFEAT) issue_panels(k0 + 32, cur ^ 1);  // overlap DMA
#pragma unroll
    for (int kk = 0; kk < 32; kk += 4) {
      int kr = kk + 2 * hi;                     // VGPR0=K{0,2}, VGPR1=K{1,3}
      v2f a;
      a.x = As[cur][(kr + 0) * LDSTR + wave * 16 + l16];
      a.y = As[cur][(kr + 1) * LDSTR + wave * 16 + l16];
#pragma unroll
      for (int s = 0; s < 4; ++s) {
        v2f bb;
        bb.x = Bs[cur][(kr + 0) * LDSTR + s * 16 + l16];
        bb.y = Bs[cur][(kr + 1) * LDSTR + s * 16 + l16];
        acc[s] = wmma4(a, bb, acc[s]);
      }
    }
    WAIT_ASYNC0();
    __syncthreads();
  }

  // epilogue: D = a2 + b2 - 2*prod into padded LDS tile
  const float* a2b = a2 + b * NPTS;
  const float* b2b = b2 + b * NPTS;
#pragma unroll
  for (int s = 0; s < 4; ++s)
#pragma unroll
    for (int v = 0; v < 8; ++v) {
      int il = wave * 16 + v + 8 * hi;          // C layout: M = v + 8*(lane>>4)
      int jl = s * 16 + l16;
      Dt[il * 65 + jl] = a2b[i0 + il] + b2b[j0 + jl] - 2.f * acc[s][v];
    }
  __syncthreads();

  if (tid < 64) {                               // row argmin over 64 j's
    float best = Dt[tid * 65];
    int bj = 0;
    for (int j = 1; j < 64; ++j) {
      float d = Dt[tid * 65 + j];
      if (d < best) { best = d; bj = j; }
    }
    unsigned long long key =
        ((unsigned long long)ordf(best) << 32) | (unsigned int)(j0 + bj);
    atomicMin(&rowmin[(size_t)b * NPTS + i0 + tid], key);
  } else {                                      // col argmin over 64 i's
    int t = tid - 64;
    float best = Dt[t];
    int bi = 0;
    for (int i = 1; i < 64; ++i) {
      float d = Dt[i * 65 + t];
      if (d < best) { best = d; bi = i; }
    }
    unsigned long long key =
        ((unsigned long long)ordf(best) << 32) | (unsigned int)(i0 + bi);
    atomicMin(&colmin[(size_t)b * NPTS + j0 + t], key);
  }
}

// ---------------------------------------------------------------------------
// Stage 3: match post-processing
// ---------------------------------------------------------------------------
__global__ void ids_back_kernel(const unsigned long long* rowmin,
                                const unsigned long long* colmin,
                                int* ids, int* back) {
  int b = blockIdx.y, i = blockIdx.x * 256 + threadIdx.x;
  ids[b * NPTS + i]  = (int)(rowmin[(size_t)b * NPTS + i] & 0xFFFFFFFFULL);
  back[b * NPTS + i] = (int)(colmin[(size_t)b * NPTS + i] & 0xFFFFFFFFULL);
}

__global__ void mask_kernel(const int* ids, const int* back, int* mask) {
  int b = blockIdx.y, i = blockIdx.x * 256 + threadIdx.x;
  int j = ids[b * NPTS + i];
  mask[b * NPTS + i] = (back[b * NPTS + j] == i) ? 1 : 0;
}

// stable compaction of valid indices (per batch), one block per batch
__global__ __launch_bounds__(256) void compact_kernel(const int* mask,
                                                      int* vlist, int* counts) {
  int b = blockIdx.x, t = threadIdx.x;
  __shared__ int cnt[256];
  __shared__ int off[256];
  int base = t * 16, c = 0;
  for (int k = 0; k < 16; ++k) c += mask[b * NPTS + base + k];
  cnt[t] = c;
  __syncthreads();
  if (t == 0) {
    int s = 0;
    for (int q = 0; q < 256; ++q) { off[q] = s; s += cnt[q]; }
    counts[b] = s;
  }
  __syncthreads();
  int o = off[t];
  for (int k = 0; k < 16; ++k) {
    int idx = base + k;
    if (mask[b * NPTS + idx]) vlist[b * NPTS + o++] = idx;
  }
}

__global__ void lmin_kernel(const int* counts, int* Lbuf) {
  int m = counts[0];
  for (int b = 1; b < BATCH; ++b) m = min(m, counts[b]);
  *Lbuf = m;
}

__global__ void select_kernel(const float* AP, const float* BP, const int* ids,
                              const int* vlist, const int* Lptr, float* x4) {
  int L = *Lptr;
  int n = blockIdx.x * 256 + threadIdx.x, b = blockIdx.y;
  if (n >= L) return;
  int idx = vlist[b * NPTS + n];
  int j   = ids[b * NPTS + idx];
  x4[((size_t)b * 4 + 0) * NPTS + n] = AP[((size_t)b * 2 + 0) * NPTS + idx];
  x4[((size_t)b * 4 + 1) * NPTS + n] = AP[((size_t)b * 2 + 1) * NPTS + idx];
  x4[((size_t)b * 4 + 2) * NPTS + n] = BP[((size_t)b * 2 + 0) * NPTS + j];
  x4[((size_t)b * 4 + 3) * NPTS + n] = BP[((size_t)b * 2 + 1) * NPTS + j];
}

// ---------------------------------------------------------------------------
// Stage 4: generic WMMA GEMM  Y[b] = A[b](MxK) * X[b](K x 4096) (+bias)
// A row-major (row stride K), optionally per-batch (a_bstride).
// X K-panels double-buffered via async DMA (zero-padded tail rows for Cin=4);
// weight panels double-buffered via guarded scalar staging.
// ---------------------------------------------------------------------------
__global__ __launch_bounds__(128) void gemm_wmma_kernel(
    const float* __restrict__ A, long a_bstride, const float* __restrict__ X,
    const float* __restrict__ bias, float* __restrict__ Y, int M, int K,
    const int* __restrict__ Lptr) {
  int L = *Lptr;
  int n0 = blockIdx.x * 64;
  if (n0 >= L) return;
  int m0 = blockIdx.y * 64;
  int b  = blockIdx.z;
  const float* Ab = A + (size_t)b * a_bstride;
  const float* Xb = X + (size_t)b * K * NPTS;

  __shared__ __align__(16) float Ws[2][32 * LDSTR];  // [k][m]
  __shared__ __align__(16) float Xs[2][32 * LDSTR];  // [k][n]

  const int tid  = threadIdx.x;
  const int lane = tid & 31;
  const int wave = tid >> 5;
  const int l16  = lane & 15;
  const int hi   = lane >> 4;

  auto issue_w = [&](int k0, int buf) {
    for (int r = 0; r < 16; ++r) {              // weights: 2048 scalars
      int idx = tid + r * 128;
      int m = idx >> 5, k = idx & 31;
      float v = 0.f;
      if (k0 + k < K && m0 + m < M) v = Ab[(size_t)(m0 + m) * K + k0 + k];
      Ws[buf][k * LDSTR + m] = v;
    }
  };
  auto issue_x = [&](int k0, int buf) {
    for (int r = 0; r < 4; ++r) {               // activations: 512 float4
      int idx = tid + r * 128;
      int row = idx >> 4;
      int c4  = (idx & 15) << 2;
      if (k0 + row < K) {
        ASYNC_B128(lds_off(&Xs[buf][row * LDSTR + c4]),
                   (unsigned long long)(Xb + (size_t)(k0 + row) * NPTS + n0 + c4));
      } else {
        *(float4*)&Xs[buf][row * LDSTR + c4] = make_float4(0.f, 0.f, 0.f, 0.f);
      }
    }
  };

  v8f acc[4];
  for (int s = 0; s < 4; ++s)
    for (int q = 0; q < 8; ++q) acc[s][q] = 0.f;

  issue_w(0, 0);
  issue_x(0, 0);
  WAIT_ASYNC0();
  __syncthreads();

  for (int k0 = 0; k0 < K; k0 += 32) {
    int cur = (k0 >> 5) & 1;
    if (k0 + 32 < K) { issue_w(k0 + 32, cur ^ 1); issue_x(k0 + 32, cur ^ 1); }
#pragma unroll
    for (int kk = 0; kk < 32; kk += 4) {
      int kr = kk + 2 * hi;
      v2f a;
      a.x = Ws[cur][(kr + 0) * LDSTR + wave * 16 + l16];
      a.y = Ws[cur][(kr + 1) * LDSTR + wave * 16 + l16];
#pragma unroll
      for (int s = 0; s < 4; ++s) {
        v2f bb;
        bb.x = Xs[cur][(kr + 0) * LDSTR + s * 16 + l16];
        bb.y = Xs[cur][(kr + 1) * LDSTR + s * 16 + l16];
        acc[s] = wmma4(a, bb, acc[s]);
      }
    }
    WAIT_ASYNC0();
    __syncthreads();
  }

#pragma unroll
  for (int s = 0; s < 4; ++s)
#pragma unroll
    for (int v = 0; v < 8; ++v) {
      int m = m0 + wave * 16 + v + 8 * hi;
      int n = n0 + s * 16 + l16;
      if (m < M) {
        float val = acc[s][v] + (bias ? bias[m] : 0.f);
        Y[((size_t)b * M + m) * NPTS + n] = val;
      }
    }
}

// ---------------------------------------------------------------------------
// BatchNorm (batch statistics over b,n<L) + ReLU, and pooling / small layers
// ---------------------------------------------------------------------------
__global__ __launch_bounds__(256) void bn_stats_kernel(const float* __restrict__ Y,
                                                       int C, const int* Lptr,
                                                       float* __restrict__ stats) {
  int c = blockIdx.x, t = threadIdx.x, L = *Lptr;
  float s = 0.f, s2 = 0.f;
  for (int b = 0; b < BATCH; ++b) {
    const float* p = Y + ((size_t)b * C + c) * NPTS;
    for (int n = t; n < L; n += 256) { float v = p[n]; s += v; s2 += v * v; }
  }
  __shared__ float r1[256], r2[256];
  r1[t] = s; r2[t] = s2;
  __syncthreads();
  for (int o = 128; o > 0; o >>= 1) {
    if (t < o) { r1[t] += r1[t + o]; r2[t] += r2[t + o]; }
    __syncthreads();
  }
  if (t == 0) { stats[2 * c] = r1[0]; stats[2 * c + 1] = r2[0]; }
}

__global__ void bn_relu_kernel(float* Y, const float* stats, const float* g,
                               const float* beta, int C, const int* Lptr) {
  int L = *Lptr;
  int n = blockIdx.x * 256 + threadIdx.x, c = blockIdx.y, b = blockIdx.z;
  if (n >= L) return;
  float cnt = 8.f * (float)L;
  float m   = stats[2 * c] / cnt;
  float var = stats[2 * c + 1] / cnt - m * m;
  float sc  = g[c] * rsqrtf(var + 1e-5f);
  size_t o  = ((size_t)b * C + c) * NPTS + n;
  float y   = (Y[o] - m) * sc + beta[c];
  Y[o] = y > 0.f ? y : 0.f;
}

__global__ __launch_bounds__(256) void maxpool_kernel(const float* __restrict__ Y,
                                                      int C, const int* Lptr,
                                                      float* __restrict__ out) {
  int c = blockIdx.x, b = blockIdx.y, t = threadIdx.x, L = *Lptr;
  const float* p = Y + ((size_t)b * C + c) * NPTS;
  float m = -3.0e38f;
  for (int n = t; n < L; n += 256) m = fmaxf(m, p[n]);
  __shared__ float r[256];
  r[t] = m;
  __syncthreads();
  for (int o = 128; o > 0; o >>= 1) {
    if (t < o) r[t] = fmaxf(r[t], r[t + o]);
    __syncthreads();
  }
  if (t == 0) out[b * C + c] = r[0];
}

__global__ void linear_kernel(const float* X, const float* W, const float* bias,
                              float* Y, int Cin, int Cout) {
  int idx = blockIdx.x * 256 + threadIdx.x;
  if (idx >= BATCH * Cout) return;
  int b = idx / Cout, o = idx % Cout;
  float s = bias ? bias[o] : 0.f;
  const float* x = X + (size_t)b * Cin;
  const float* w = W + (size_t)o * Cin;
  for (int i = 0; i < Cin; ++i) s += x[i] * w[i];
  Y[idx] = s;
}

__global__ void bn_batch_relu_kernel(float* Y, const float* g, const float* beta,
                                     int C) {
  int c = blockIdx.x * 256 + threadIdx.x;
  if (c >= C) return;
  float s = 0.f, s2 = 0.f;
  for (int b = 0; b < BATCH; ++b) { float v = Y[b * C + c]; s += v; s2 += v * v; }
  float m = s / 8.f, var = s2 / 8.f - m * m;
  float sc = g[c] * rsqrtf(var + 1e-5f);
  for (int b = 0; b < BATCH; ++b) {
    float y = (Y[b * C + c] - m) * sc + beta[c];
    Y[b * C + c] = y > 0.f ? y : 0.f;
  }
}

__global__ void add_eye_kernel(float* T, int K) {
  int idx = blockIdx.x * 256 + threadIdx.x;
  if (idx >= BATCH * K * K) return;
  int r = idx % (K * K);
  if (r / K == r % K) T[idx] += 1.0f;
}

__global__ void apply_t1_kernel(float* x4, const float* T1, const int* Lptr) {
  int L = *Lptr;
  int n = blockIdx.x * 256 + threadIdx.x, b = blockIdx.y;
  if (n >= L) return;
  float v[4], o[4];
  for (int j = 0; j < 4; ++j) v[j] = x4[((size_t)b * 4 + j) * NPTS + n];
  for (int i = 0; i < 4; ++i) {
    float s = 0.f;
    for (int j = 0; j < 4; ++j) s += T1[b * 16 + i * 4 + j] * v[j];
    o[i] = s;
  }
  for (int i = 0; i < 4; ++i) x4[((size_t)b * 4 + i) * NPTS + n] = o[i];
}

__global__ void cat_kernel(const float* le, const float* gf, float* cat,
                           const int* Lptr) {
  int L = *Lptr;
  int n = blockIdx.x * 256 + threadIdx.x, c = blockIdx.y, b = blockIdx.z;
  if (n >= L) return;
  float v = (c < 64) ? le[((size_t)b * 64 + c) * NPTS + n]
                     : gf[b * 1024 + (c - 64)];
  cat[((size_t)b * 1088 + c) * NPTS + n] = v;
}

__global__ void head_kernel(const float* H, const float* W, const float* bias,
                            float* w, const int* Lptr) {
  int L = *Lptr;
  int n = blockIdx.x * 256 + threadIdx.x, b = blockIdx.y;
  if (n >= L) return;
  float s = bias[0];
  for (int c = 0; c < 128; ++c) s += W[c] * H[((size_t)b * 128 + c) * NPTS + n];
  w[b * NPTS + n] = 1.f / (1.f + expf(-s));
}

// outputs: x=(8,64,L), w=(8,L), T2=(8,64,64) concatenated flat
__global__ void out_x_w_kernel(const float* x64, const float* wv, float* out,
                               const int* Lptr) {
  int L = *Lptr;
  int n = blockIdx.x * 256 + threadIdx.x;
  if (n >= L) return;
  int c = blockIdx.y, b = blockIdx.z;
  if (c < 64)
    out[((size_t)b * 64 + c) * L + n] = x64[((size_t)b * 64 + c) * NPTS + n];
  else
    out[(size_t)BATCH * 64 * L + (size_t)b * L + n] = wv[b * NPTS + n];
}

__global__ void out_t2_kernel(const float* T2, float* out, const int* Lptr) {
  int L = *Lptr;
  int idx = blockIdx.x * 256 + threadIdx.x;
  if (idx >= BATCH * 64 * 64) return;
  out[(size_t)BATCH * 64 * L + (size_t)BATCH * L + idx] = T2[idx];
}

// ---------------------------------------------------------------------------
// Host orchestration
// ---------------------------------------------------------------------------
extern "C" void kernel_launch(void* const* d_in, const int* in_sizes, int n_in,
                              void* d_out, int out_size, void* d_ws,
                              size_t ws_size, hipStream_t stream) {
  (void)in_sizes; (void)n_in; (void)out_size; (void)ws_size;
  const float* AP = (const float*)d_in[0];
  const float* AF = (const float*)d_in[1];
  const float* BP = (const float*)d_in[2];
  const float* BF = (const float*)d_in[3];
  // params flattened (jax tree order: sorted dict keys, lists in order):
  // binseg[4x{W,b,beta,g}]=4.., embedding_t{blocks[3x4],bn1{beta,g},bn2{beta,g},
  // l1{W,b},l2{W,b},l3{W,b}}=20.., head{W,b}=42.., input_t=44.., mlp1=66..,
  // mlp2=74..
  auto F = [&](int i) { return (const float*)d_in[i]; };

  // workspace carve-up
  char* ws = (char*)d_ws;
  size_t off = 0;
  auto alloc = [&](size_t bytes) {
    void* p = ws + off;
    off += (bytes + 255) & ~(size_t)255;
    return p;
  };
  unsigned long long* rowmin = (unsigned long long*)alloc(BATCH * NPTS * 8);
  unsigned long long* colmin = (unsigned long long*)alloc(BATCH * NPTS * 8);
  float* a2    = (float*)alloc(BATCH * NPTS * 4);
  float* b2    = (float*)alloc(BATCH * NPTS * 4);
  int*   ids   = (int*)alloc(BATCH * NPTS * 4);
  int*   back  = (int*)alloc(BATCH * NPTS * 4);
  int*   maskb = (int*)alloc(BATCH * NPTS * 4);
  int*   vlist = (int*)alloc(BATCH * NPTS * 4);
  int*   counts= (int*)alloc(BATCH * 4);
  int*   Lptr  = (int*)alloc(4);
  float* x4    = (float*)alloc((size_t)BATCH * 4 * NPTS * 4);
  float* xle   = (float*)alloc((size_t)BATCH * 64 * NPTS * 4);
  float* lebuf = (float*)alloc((size_t)BATCH * 64 * NPTS * 4);
  float* gfbuf = (float*)alloc(BATCH * 1024 * 4);
  float* fcA   = (float*)alloc(BATCH * 512 * 4);
  float* fcB   = (float*)alloc(BATCH * 256 * 4);
  float* T1buf = (float*)alloc(BATCH * 16 * 4);
  float* T2buf = (float*)alloc(BATCH * 64 * 64 * 4);
  float* stats = (float*)alloc(2 * 1088 * 4);
  float* wbuf  = (float*)alloc(BATCH * NPTS * 4);
  float* actA  = (float*)alloc((size_t)BATCH * 1088 * NPTS * 4);
  float* actB  = (float*)alloc((size_t)BATCH * 1024 * NPTS * 4);

  dim3 bn256(256);
  dim3 gBN(NPTS / 256, BATCH);

  auto gemm = [&](const float* A, long abstr, const float* X, const float* bias,
                  float* Y, int M, int K) {
    dim3 g(NPTS / 64, (M + 63) / 64, BATCH);
    gemm_wmma_kernel<<<g, 128, 0, stream>>>(A, abstr, X, bias, Y, M, K, Lptr);
  };
  auto cbr = [&](const float* in, float* out, int p, int Cin, int Cout) {
    gemm(F(p + 0), 0, in, F(p + 1), out, Cout, Cin);
    bn_stats_kernel<<<Cout, 256, 0, stream>>>(out, Cout, Lptr, stats);
    dim3 g2(NPTS / 256, Cout, BATCH);
    bn_relu_kernel<<<g2, 256, 0, stream>>>(out, stats, F(p + 3), F(p + 2), Cout,
                                           Lptr);
  };
  auto lbr = [&](const float* in, float* out, int pl, int pbn, int Ci, int Co) {
    linear_kernel<<<(BATCH * Co + 255) / 256, 256, 0, stream>>>(in, F(pl),
                                                                F(pl + 1), out,
                                                                Ci, Co);
    bn_batch_relu_kernel<<<(Co + 255) / 256, 256, 0, stream>>>(out, F(pbn + 1),
                                                               F(pbn), Co);
  };

  // ---- mutual-NN matching ----
  init_u64_kernel<<<(BATCH * NPTS + 255) / 256, bn256, 0, stream>>>(rowmin,
                                                                    BATCH * NPTS);
  init_u64_kernel<<<(BATCH * NPTS + 255) / 256, bn256, 0, stream>>>(colmin,
                                                                    BATCH * NPTS);
  norms_kernel<<<gBN, bn256, 0, stream>>>(AF, a2);
  norms_kernel<<<gBN, bn256, 0, stream>>>(BF, b2);
  match_kernel<<<dim3(NPTS / 64, NPTS / 64, BATCH), 128, 0, stream>>>(
      AF, BF, a2, b2, rowmin, colmin);
  ids_back_kernel<<<gBN, bn256, 0, stream>>>(rowmin, colmin, ids, back);
  mask_kernel<<<gBN, bn256, 0, stream>>>(ids, back, maskb);
  compact_kernel<<<BATCH, 256, 0, stream>>>(maskb, vlist, counts);
  lmin_kernel<<<1, 1, 0, stream>>>(counts, Lptr);
  select_kernel<<<gBN, bn256, 0, stream>>>(AP, BP, ids, vlist, Lptr, x4);

  // ---- input TNet (K=4): params at 44.. ----
  cbr(x4, actA, 44, 4, 64);
  cbr(actA, actB, 48, 64, 128);
  cbr(actB, actA, 52, 128, 1024);
  maxpool_kernel<<<dim3(1024, BATCH), 256, 0, stream>>>(actA, 1024, Lptr, gfbuf);
  lbr(gfbuf, fcA, 60, 56, 1024, 512);
  lbr(fcA, fcB, 62, 58, 512, 256);
  linear_kernel<<<(BATCH * 16 + 255) / 256, 256, 0, stream>>>(fcB, F(64), F(65),
                                                              T1buf, 256, 16);
  add_eye_kernel<<<1, 256, 0, stream>>>(T1buf, 4);
  apply_t1_kernel<<<gBN, bn256, 0, stream>>>(x4, T1buf, Lptr);

  // ---- mlp1: 66.. -> xle (returned as output x) ----
  cbr(x4, actA, 66, 4, 64);
  cbr(actA, xle, 70, 64, 64);

  // ---- embedding TNet (K=64): 20.. ----
  cbr(xle, actA, 20, 64, 64);
  cbr(actA, actB, 24, 64, 128);
  cbr(actB, actA, 28, 128, 1024);
  maxpool_kernel<<<dim3(1024, BATCH), 256, 0, stream>>>(actA, 1024, Lptr, gfbuf);
  lbr(gfbuf, fcA, 36, 32, 1024, 512);
  lbr(fcA, fcB, 38, 34, 512, 256);
  linear_kernel<<<(BATCH * 4096 + 255) / 256, 256, 0, stream>>>(fcB, F(40),
                                                                F(41), T2buf,
                                                                256, 4096);
  add_eye_kernel<<<(BATCH * 4096 + 255) / 256, 256, 0, stream>>>(T2buf, 64);

  // ---- le = T2 @ x (per-batch A) ----
  gemm(T2buf, 64 * 64, xle, nullptr, lebuf, 64, 64);

  // ---- mlp2: 74.. ; global feature ----
  cbr(lebuf, actA, 74, 64, 64);
  cbr(actA, actB, 78, 64, 128);
  cbr(actB, actA, 82, 128, 1024);
  maxpool_kernel<<<dim3(1024, BATCH), 256, 0, stream>>>(actA, 1024, Lptr, gfbuf);

  // ---- concat + binseg (4..) + head (42..) ----
  cat_kernel<<<dim3(NPTS / 256, 1088, BATCH), 256, 0, stream>>>(lebuf, gfbuf,
                                                                actA, Lptr);
  cbr(actA, actB, 4, 1088, 512);
  cbr(actB, actA, 8, 512, 256);
  cbr(actA, actB, 12, 256, 128);
  cbr(actB, actA, 16, 128, 128);
  head_kernel<<<gBN, bn256, 0, stream>>>(actA, F(42), F(43), wbuf, Lptr);

  // ---- pack outputs ----
  out_x_w_kernel<<<dim3(NPTS / 256, 65, BATCH), 256, 0, stream>>>(
      xle, wbuf, (float*)d_out, Lptr);
  out_t2_kernel<<<(BATCH * 64 * 64 + 255) / 256, 256, 0, stream>>>(
      T2buf, (float*)d_out, Lptr);
}